// L1Loss_82978768159400
// MI455X (gfx1250) — compile-verified
//
#include <hip/hip_runtime.h>
#include <hip/hip_bf16.h>
#include <stdint.h>

// ---------------------------------------------------------------------------
// GraphSpmv + masked L1 loss for MI455X (gfx1250).
// Bandwidth-bound: 320MB edge stream @ 23.3 TB/s ~= 13.7us floor.
// d / Ad (2MB each) are L2-resident (192MB L2), so gather+atomic scatter are cheap.
// CDNA5 paths: global_load_async_to_lds_b128 (NT) double buffering (ASYNCcnt),
// ds_load_b128 consumption, native global_atomic_add_f32 scatter at DEV scope.
// ---------------------------------------------------------------------------

#define SPMV_BLOCK 256
#define EDGES_PER_THREAD 4
#define TILE_EDGES (SPMV_BLOCK * EDGES_PER_THREAD)   // 1024 edges per block-tile

// LDS layout (double buffered):
//   vals : [0,      8192)  : 16B per (buf,slot)
//   dst  : [8192,  24576)  : 32B per (buf,slot)
//   src  : [24576, 40960)  : 32B per (buf,slot)
#define LDS_BYTES 40960
#define LDS_VOFF  0
#define LDS_DOFF  8192
#define LDS_SOFF  24576

// Native no-return f32 atomic add at device scope (visible to following kernels
// through L2; STOREcnt-tracked, drained by s_endpgm's implicit wait-idle).
__device__ __forceinline__ void atom_add_f32_dev(float* p, float v) {
    asm volatile("global_atomic_add_f32 %0, %1, off scope:SCOPE_DEV"
                 :: "v"(p), "v"(v) : "memory");
}

// Issue 5 async b128 copies (4 vals, 4 dst i64, 4 src i64) into LDS buffer `buf`.
// Non-temporal: edge streams are touch-once; keep L2 for d/Ad.
__device__ __forceinline__ void prefetch_tile(uint32_t sbase, int buf, long long tile,
                                              const float* __restrict__ vals,
                                              const long long* __restrict__ dstp,
                                              const long long* __restrict__ srcp,
                                              long long E) {
    long long base = tile * (long long)TILE_EDGES + (long long)threadIdx.x * EDGES_PER_THREAD;
    if (base > E - EDGES_PER_THREAD) base = E - EDGES_PER_THREAD;  // clamp (E multiple of 4)
    uint32_t slot   = (uint32_t)(buf * SPMV_BLOCK + (int)threadIdx.x);
    uint32_t v_lds  = sbase + LDS_VOFF + slot * 16u;
    uint32_t di_lds = sbase + LDS_DOFF + slot * 32u;
    uint32_t si_lds = sbase + LDS_SOFF + slot * 32u;
    const float*     vg = vals + base;
    const long long* dg = dstp + base;
    const long long* sg = srcp + base;
    // Make sure prior ds_loads of this buffer have completed before overwrite.
    asm volatile("s_wait_dscnt 0x0" ::: "memory");
    asm volatile("global_load_async_to_lds_b128 %0, %1, off th:TH_LOAD_NT"
                 :: "v"(v_lds), "v"(vg) : "memory");
    asm volatile("global_load_async_to_lds_b128 %0, %1, off th:TH_LOAD_NT"
                 :: "v"(di_lds), "v"(dg) : "memory");
    asm volatile("global_load_async_to_lds_b128 %0, %1, off offset:16 th:TH_LOAD_NT"
                 :: "v"(di_lds), "v"(dg) : "memory");
    asm volatile("global_load_async_to_lds_b128 %0, %1, off th:TH_LOAD_NT"
                 :: "v"(si_lds), "v"(sg) : "memory");
    asm volatile("global_load_async_to_lds_b128 %0, %1, off offset:16 th:TH_LOAD_NT"
                 :: "v"(si_lds), "v"(sg) : "memory");
}

__global__ void __launch_bounds__(SPMV_BLOCK)
spmv_edges_kernel(const long long* __restrict__ edge_index,  // [2*E]: dst then src
                  const float* __restrict__ vals,            // [E]
                  const float* __restrict__ dvec,            // [N]
                  float* __restrict__ Ad,                    // [N] (pre-zeroed)
                  long long E) {
    __shared__ unsigned char smem[LDS_BYTES] __attribute__((aligned(16)));
    const long long* dstp = edge_index;
    const long long* srcp = edge_index + E;
    const long long tiles = (E + TILE_EDGES - 1) / TILE_EDGES;

    // Low 32 bits of the generic address of an LDS object == LDS byte offset.
    uint32_t sbase = (uint32_t)(uintptr_t)(&smem[0]);

    long long tile = (long long)blockIdx.x;
    if (tile >= tiles) return;

    int buf = 0;
    prefetch_tile(sbase, buf, tile, vals, dstp, srcp, E);

    for (; tile < tiles; tile += (long long)gridDim.x) {
        long long next = tile + (long long)gridDim.x;
        if (next < tiles) {
            prefetch_tile(sbase, buf ^ 1, next, vals, dstp, srcp, E);
            asm volatile("s_wait_asynccnt 0x5" ::: "memory");  // current buf ready
        } else {
            asm volatile("s_wait_asynccnt 0x0" ::: "memory");
        }

        // Consume this thread's 4 edges from LDS (ds_load_b128s).
        uint32_t slot = (uint32_t)(buf * SPMV_BLOCK + (int)threadIdx.x);
        const float4*     vv = reinterpret_cast<const float4*>(smem + LDS_VOFF + (size_t)slot * 16u);
        const longlong2*  dd = reinterpret_cast<const longlong2*>(smem + LDS_DOFF + (size_t)slot * 32u);
        const longlong2*  ss = reinterpret_cast<const longlong2*>(smem + LDS_SOFF + (size_t)slot * 32u);
        float4    v   = *vv;
        longlong2 d01 = dd[0];
        longlong2 d23 = dd[1];
        longlong2 s01 = ss[0];
        longlong2 s23 = ss[1];

        long long b0 = tile * (long long)TILE_EDGES + (long long)threadIdx.x * EDGES_PER_THREAD;
        if (b0 + EDGES_PER_THREAD <= E) {
            // Gather from L2-resident d, scatter via native f32 atomics.
            float c0 = v.x * dvec[(int)s01.x];
            float c1 = v.y * dvec[(int)s01.y];
            float c2 = v.z * dvec[(int)s23.x];
            float c3 = v.w * dvec[(int)s23.y];
            atom_add_f32_dev(&Ad[(int)d01.x], c0);
            atom_add_f32_dev(&Ad[(int)d01.y], c1);
            atom_add_f32_dev(&Ad[(int)d23.x], c2);
            atom_add_f32_dev(&Ad[(int)d23.y], c3);
        } else if (b0 < E) {
            float     cv[4] = {v.x, v.y, v.z, v.w};
            long long sv[4] = {s01.x, s01.y, s23.x, s23.y};
            long long dv[4] = {d01.x, d01.y, d23.x, d23.y};
            for (int j = 0; j < EDGES_PER_THREAD; ++j) {
                if (b0 + j < E) {
                    atom_add_f32_dev(&Ad[(int)dv[j]], cv[j] * dvec[(int)sv[j]]);
                }
            }
        }
        buf ^= 1;
    }
}

__global__ void init_kernel(float* __restrict__ Ad, float* __restrict__ out, int N) {
    int i = blockIdx.x * blockDim.x + threadIdx.x;
    if (i < N) Ad[i] = 0.0f;
    if (i == 0) out[0] = 0.0f;
}

__global__ void __launch_bounds__(256)
loss_kernel(const float* __restrict__ Ad,
            const unsigned char* __restrict__ mask,
            const float* __restrict__ res,
            float* __restrict__ out, int N, float invN) {
    float acc = 0.0f;
    for (int i = blockIdx.x * blockDim.x + threadIdx.x; i < N;
         i += gridDim.x * blockDim.x) {
        float a = mask[i] ? Ad[i] : 0.0f;
        acc += fabsf(a - res[i]);
    }
    // wave32 shuffle reduce
    for (int off = 16; off > 0; off >>= 1)
        acc += __shfl_down(acc, off, 32);
    __shared__ float ws[8];
    int lane = threadIdx.x & 31;
    int wave = threadIdx.x >> 5;
    if (lane == 0) ws[wave] = acc;
    __syncthreads();
    if (wave == 0) {
        float b = (lane < (int)(blockDim.x >> 5)) ? ws[lane] : 0.0f;
        for (int off = 4; off > 0; off >>= 1)
            b += __shfl_down(b, off, 32);
        if (lane == 0)
            atom_add_f32_dev(out, b * invN);
    }
}

extern "C" void kernel_launch(void* const* d_in, const int* in_sizes, int n_in,
                              void* d_out, int out_size, void* d_ws, size_t ws_size,
                              hipStream_t stream) {
    const float*         dvec = (const float*)d_in[0];          // [N] f32
    const long long*     ei   = (const long long*)d_in[1];      // [2*E] i64
    const float*         vals = (const float*)d_in[2];          // [E] f32
    const unsigned char* mask = (const unsigned char*)d_in[3];  // [N] bool
    const float*         res  = (const float*)d_in[4];          // [N] f32

    int       N = in_sizes[0];
    long long E = (long long)in_sizes[1] / 2;

    float* Ad  = (float*)d_ws;
    float* out = (float*)d_out;

    // 1) zero accumulators (must happen every call)
    int initBlocks = (N + 255) / 256;
    init_kernel<<<initBlocks, 256, 0, stream>>>(Ad, out, N);

    // 2) SpMV scatter
    long long tiles = (E + TILE_EDGES - 1) / TILE_EDGES;
    long long gb = tiles < 2048 ? tiles : 2048;
    spmv_edges_kernel<<<(int)gb, SPMV_BLOCK, 0, stream>>>(ei, vals, dvec, Ad, E);

    // 3) masked L1 mean
    loss_kernel<<<512, 256, 0, stream>>>(Ad, mask, res, out, N, 1.0f / (float)N);
}